// A2C_DND_LSTM_26774826123372
// MI455X (gfx1250) — compile-verified
//
#include <hip/hip_runtime.h>
#include <hip/hip_bf16.h>
#include <math.h>

// ---------------------------------------------------------------------------
// A2C DND-LSTM step for MI455X (gfx1250, wave32).
// Bandwidth-bound: 410MB of mem_keys/mem_vals per call -> ~17.6us @ 23.3TB/s.
// fp32 throughout. DND scan kernels use the CDNA5 async-to-LDS path
// (global_load_async_to_lds_b128 + s_wait_asynccnt) with per-wave double
// buffering; Wh@h0 matvec uses v_wmma_f32_16x16x4_f32.
// ---------------------------------------------------------------------------

#define HDIM   512
#define G1     2048   // blocks for key scan
#define G3     512    // blocks for value scan
#define BLK    256    // 8 waves per block
#define WAVES  8
#define NROWS5H 2560  // 5*H rows of preact

typedef float v2f __attribute__((ext_vector_type(2)));
typedef float v8f __attribute__((ext_vector_type(8)));

// Stage one 512-float row (2KB) into LDS: 4 async B128 copies, 512B each
// (32 lanes x 16B). INST_OFFSET is added to BOTH global and LDS addresses
// (ISA 15.18.3), so one LDS-addr VGPR + one global-addr pair suffice.
__device__ __forceinline__ void async_stage_row(const float* g, unsigned lds) {
  asm volatile("global_load_async_to_lds_b128 %0, %1, off"
               :: "v"(lds), "v"(g) : "memory");
  asm volatile("global_load_async_to_lds_b128 %0, %1, off offset:512"
               :: "v"(lds), "v"(g) : "memory");
  asm volatile("global_load_async_to_lds_b128 %0, %1, off offset:1024"
               :: "v"(lds), "v"(g) : "memory");
  asm volatile("global_load_async_to_lds_b128 %0, %1, off offset:1536"
               :: "v"(lds), "v"(g) : "memory");
}

// ---------------- Kernel 1: sims[n] = -||K[n]-cue||^2, + per-block max -----
__global__ __launch_bounds__(BLK)
void k_sims(const float* __restrict__ K, const float* __restrict__ cue,
            float* __restrict__ sims, float* __restrict__ bmax, int N) {
  __shared__ __align__(16) float scue[HDIM];
  __shared__ __align__(16) float stage[WAVES][2][HDIM];  // 32KB double buffer
  __shared__ float wmax[WAVES];
  for (int i = threadIdx.x; i < HDIM; i += BLK) scue[i] = cue[i];
  __syncthreads();

  const int lane = threadIdx.x & 31;
  const int wave = threadIdx.x >> 5;
  const int gw   = blockIdx.x * WAVES + wave;
  const int tw   = gridDim.x * WAVES;

  // Per-lane LDS byte addresses of the two staging slots (low 32 bits of the
  // generic pointer are the LDS offset on gfx1250).
  const unsigned lbase0 = (unsigned)(uintptr_t)&stage[wave][0][lane * 4];
  const unsigned lbase1 = (unsigned)(uintptr_t)&stage[wave][1][lane * 4];

  float lmax = -INFINITY;
  int buf = 0;
  if (gw < N) async_stage_row(K + (size_t)gw * HDIM + lane * 4, lbase0);
  for (int row = gw; row < N; row += tw) {
    const int nxt = row + tw;
    if (nxt < N) {  // issue next row into the other buffer, wait for current
      async_stage_row(K + (size_t)nxt * HDIM + lane * 4, buf ? lbase0 : lbase1);
      if (nxt + tw < N)  // warm L2 two rows ahead (global_prefetch_b8)
        __builtin_prefetch(K + (size_t)(nxt + tw) * HDIM + lane * 4, 0, 0);
      asm volatile("s_wait_asynccnt 0x4" ::: "memory");
    } else {
      asm volatile("s_wait_asynccnt 0x0" ::: "memory");
    }
    const float* sr = &stage[wave][buf][0];
    float s = 0.f;
#pragma unroll
    for (int i = 0; i < 4; ++i) {
      const int j = lane * 4 + i * 128;
      float4 kv = *reinterpret_cast<const float4*>(sr + j);
      float4 cv = *reinterpret_cast<const float4*>(&scue[j]);
      float dx = kv.x - cv.x; s = fmaf(dx, dx, s);
      float dy = kv.y - cv.y; s = fmaf(dy, dy, s);
      float dz = kv.z - cv.z; s = fmaf(dz, dz, s);
      float dw = kv.w - cv.w; s = fmaf(dw, dw, s);
    }
#pragma unroll
    for (int off = 16; off; off >>= 1) s += __shfl_xor(s, off, 32);
    if (lane == 0) { float v = -s; sims[row] = v; lmax = fmaxf(lmax, v); }
    buf ^= 1;
  }
#pragma unroll
  for (int off = 16; off; off >>= 1) lmax = fmaxf(lmax, __shfl_xor(lmax, off, 32));
  if (lane == 0) wmax[wave] = lmax;
  __syncthreads();
  if (threadIdx.x == 0) {
    float m = wmax[0];
#pragma unroll
    for (int w = 1; w < WAVES; ++w) m = fmaxf(m, wmax[w]);
    bmax[blockIdx.x] = m;
  }
}

// ---------------- Kernel 2: global max of per-block maxima -----------------
__global__ __launch_bounds__(BLK)
void k_maxreduce(const float* __restrict__ bmax, float* __restrict__ gmax, int n) {
  __shared__ float sm[WAVES];
  float m = -INFINITY;
  for (int i = threadIdx.x; i < n; i += BLK) m = fmaxf(m, bmax[i]);
#pragma unroll
  for (int off = 16; off; off >>= 1) m = fmaxf(m, __shfl_xor(m, off, 32));
  const int lane = threadIdx.x & 31, wave = threadIdx.x >> 5;
  if (lane == 0) sm[wave] = m;
  __syncthreads();
  if (threadIdx.x == 0) {
    float r = sm[0];
#pragma unroll
    for (int w = 1; w < WAVES; ++w) r = fmaxf(r, sm[w]);
    *gmax = r;
  }
}

// ---------------- Kernel 3: partial sums of exp(sims-max)*V ----------------
// Async-staged V rows; each lane keeps 16 private column accumulators (fixed
// column ownership), folded through LDS at block end (no float atomics).
__global__ __launch_bounds__(BLK)
void k_vals(const float* __restrict__ V, const float* __restrict__ sims,
            const float* __restrict__ gmaxp, float* __restrict__ partial, int N) {
  __shared__ __align__(16) float stage[WAVES][2][HDIM];  // 32KB double buffer
  __shared__ float wacc[WAVES][HDIM + 1];
  const int lane = threadIdx.x & 31;
  const int wave = threadIdx.x >> 5;
  const int gw   = blockIdx.x * WAVES + wave;
  const int tw   = gridDim.x * WAVES;
  const float gmax = *gmaxp;

  const unsigned lbase0 = (unsigned)(uintptr_t)&stage[wave][0][lane * 4];
  const unsigned lbase1 = (unsigned)(uintptr_t)&stage[wave][1][lane * 4];

  float4 a0 = {0,0,0,0}, a1 = {0,0,0,0}, a2 = {0,0,0,0}, a3 = {0,0,0,0};
  float esum = 0.f;
  int buf = 0;
  if (gw < N) async_stage_row(V + (size_t)gw * HDIM + lane * 4, lbase0);
  for (int row = gw; row < N; row += tw) {
    const int nxt = row + tw;
    if (nxt < N) {
      async_stage_row(V + (size_t)nxt * HDIM + lane * 4, buf ? lbase0 : lbase1);
      if (nxt + tw < N)
        __builtin_prefetch(V + (size_t)(nxt + tw) * HDIM + lane * 4, 0, 0);
      asm volatile("s_wait_asynccnt 0x4" ::: "memory");
    } else {
      asm volatile("s_wait_asynccnt 0x0" ::: "memory");
    }
    const float e = __expf(sims[row] - gmax);
    if (lane == 0) esum += e;
    const float* vr = &stage[wave][buf][0];
    float4 v0 = *reinterpret_cast<const float4*>(vr + lane * 4);
    float4 v1 = *reinterpret_cast<const float4*>(vr + lane * 4 + 128);
    float4 v2 = *reinterpret_cast<const float4*>(vr + lane * 4 + 256);
    float4 v3 = *reinterpret_cast<const float4*>(vr + lane * 4 + 384);
    a0.x = fmaf(e, v0.x, a0.x); a0.y = fmaf(e, v0.y, a0.y);
    a0.z = fmaf(e, v0.z, a0.z); a0.w = fmaf(e, v0.w, a0.w);
    a1.x = fmaf(e, v1.x, a1.x); a1.y = fmaf(e, v1.y, a1.y);
    a1.z = fmaf(e, v1.z, a1.z); a1.w = fmaf(e, v1.w, a1.w);
    a2.x = fmaf(e, v2.x, a2.x); a2.y = fmaf(e, v2.y, a2.y);
    a2.z = fmaf(e, v2.z, a2.z); a2.w = fmaf(e, v2.w, a2.w);
    a3.x = fmaf(e, v3.x, a3.x); a3.y = fmaf(e, v3.y, a3.y);
    a3.z = fmaf(e, v3.z, a3.z); a3.w = fmaf(e, v3.w, a3.w);
    buf ^= 1;
  }
  float* wa = wacc[wave];
  const int j = lane * 4;
  wa[j      ] = a0.x; wa[j + 1      ] = a0.y; wa[j + 2      ] = a0.z; wa[j + 3      ] = a0.w;
  wa[j + 128] = a1.x; wa[j + 1 + 128] = a1.y; wa[j + 2 + 128] = a1.z; wa[j + 3 + 128] = a1.w;
  wa[j + 256] = a2.x; wa[j + 1 + 256] = a2.y; wa[j + 2 + 256] = a2.z; wa[j + 3 + 256] = a2.w;
  wa[j + 384] = a3.x; wa[j + 1 + 384] = a3.y; wa[j + 2 + 384] = a3.z; wa[j + 3 + 384] = a3.w;
  if (lane == 0) wa[HDIM] = esum;
  __syncthreads();
  for (int c = threadIdx.x; c < HDIM + 1; c += BLK) {
    float s = 0.f;
#pragma unroll
    for (int w = 0; w < WAVES; ++w) s += wacc[w][c];
    partial[(size_t)blockIdx.x * (HDIM + 1) + c] = s;
  }
}

// ---------------- Kernel 4: reduce partials -> m_unnorm[512] + denom -------
__global__ __launch_bounds__(BLK)
void k_mreduce(const float* __restrict__ partial, float* __restrict__ mout, int nb) {
  for (int c = threadIdx.x; c < HDIM + 1; c += BLK) {
    float s = 0.f;
    for (int b = 0; b < nb; ++b) s += partial[(size_t)b * (HDIM + 1) + c];
    mout[c] = s;
  }
}

// ---------------- Kernel 5: preact = Wh@h0 (WMMA f32 16x16x4) + Wx@x + b ---
// One wave owns a 16-row tile. B operand broadcasts h0 across all 16 N
// columns, so every column of D is the matvec result for its M row.
__global__ __launch_bounds__(BLK)
void k_preact(const float* __restrict__ Wh, const float* __restrict__ Wx,
              const float* __restrict__ bias, const float* __restrict__ h0,
              const float* __restrict__ state, const float* __restrict__ p_action,
              const float* __restrict__ p_reward, const float* __restrict__ timestep,
              float* __restrict__ preact) {
  __shared__ __align__(16) float sh0[HDIM];
  __shared__ float sx[70];
  const int t = threadIdx.x;
  for (int i = t; i < HDIM; i += BLK) sh0[i] = h0[i];
  if (t < 64)       sx[t] = state[t];
  else if (t < 68)  sx[t] = p_action[t - 64];
  else if (t == 68) sx[68] = p_reward[0];
  else if (t == 69) sx[69] = timestep[0];
  __syncthreads();

  const int lane  = t & 31;
  const int wave  = t >> 5;
  const int tile  = blockIdx.x * WAVES + wave;      // 160 tiles total
  const int r0    = tile * 16;
  if (r0 >= NROWS5H) return;                        // wave-uniform exit

  const int m     = lane & 15;                      // M row within tile
  const int khalf = (lane >> 4) << 1;               // lanes 16-31 carry K+2,K+3

  v8f c = {0.f, 0.f, 0.f, 0.f, 0.f, 0.f, 0.f, 0.f};
  const float* wrow = Wh + (size_t)(r0 + m) * HDIM;
  for (int k = 0; k < HDIM; k += 4) {
    const int kk = k + khalf;
    v2f a;  a.x  = wrow[kk];  a.y  = wrow[kk + 1];  // A: 16x4 f32 tile
    v2f bb; bb.x = sh0[kk];   bb.y = sh0[kk + 1];   // B: 4x16, h0 broadcast
    c = __builtin_amdgcn_wmma_f32_16x16x4_f32(
        /*neg_a=*/false, a, /*neg_b=*/false, bb,
        /*c_mod=*/(short)0, c, /*reuse_a=*/false, /*reuse_b=*/false);
  }

  // Wx @ x_t contribution (70 cols): lane L computes row r0 + (L&15).
  float wx = 0.f;
  const float* xrow = Wx + (size_t)(r0 + m) * 70;
  for (int jj = 0; jj < 70; ++jj) wx = fmaf(xrow[jj], sx[jj], wx);

  // D layout: VGPR v -> lane0 holds row r0+v, lane16 holds row r0+8+v.
#pragma unroll
  for (int v = 0; v < 8; ++v) {
    const float wlo = __shfl(wx, v,     32);
    const float whi = __shfl(wx, 8 + v, 32);
    if (lane == 0)  preact[r0 + v]     = c[v] + wlo + bias[r0 + v];
    if (lane == 16) preact[r0 + 8 + v] = c[v] + whi + bias[r0 + 8 + v];
  }
}

// ---------------- Kernel 6: gates, c_t/h_t, actor softmax, critic ----------
__global__ __launch_bounds__(HDIM)
void k_final(const float* __restrict__ preact, const float* __restrict__ mun,
             const float* __restrict__ c0,
             const float* __restrict__ actor_w, const float* __restrict__ actor_b,
             const float* __restrict__ critic_w, const float* __restrict__ critic_b,
             float* __restrict__ out) {
  __shared__ float sh[HDIM];
  __shared__ float slog[5];
  const int t = threadIdx.x;                 // 512 threads
  const float inv_den = 1.0f / mun[HDIM];
  {
    const float pf = preact[t];
    const float pi = preact[HDIM + t];
    const float po = preact[2 * HDIM + t];
    const float pr = preact[3 * HDIM + t];
    const float pc = preact[4 * HDIM + t];
    const float f  = 1.f / (1.f + __expf(-pf));
    const float ii = 1.f / (1.f + __expf(-pi));
    const float o  = 1.f / (1.f + __expf(-po));
    const float r  = 1.f / (1.f + __expf(-pr));
    const float ch = tanhf(pc);
    const float mm = mun[t] * inv_den;
    const float ct = f * c0[t] + ii * ch + r * mm;
    const float ht = o * tanhf(ct);
    sh[t] = ht;
    out[5 + t]        = ht;   // h_t
    out[5 + HDIM + t] = ct;   // c_t
  }
  __syncthreads();
  const int lane = t & 31, wave = t >> 5;
  if (wave < 4) {            // actor logits, one wave per row
    const float* ar = actor_w + wave * HDIM;
    float s = 0.f;
    for (int jj = lane; jj < HDIM; jj += 32) s = fmaf(ar[jj], sh[jj], s);
#pragma unroll
    for (int off = 16; off; off >>= 1) s += __shfl_xor(s, off, 32);
    if (lane == 0) slog[wave] = s + actor_b[wave];
  } else if (wave == 4) {    // critic
    float s = 0.f;
    for (int jj = lane; jj < HDIM; jj += 32) s = fmaf(critic_w[jj], sh[jj], s);
#pragma unroll
    for (int off = 16; off; off >>= 1) s += __shfl_xor(s, off, 32);
    if (lane == 0) { const float v = s + critic_b[0]; slog[4] = v; out[4] = v; }
  }
  __syncthreads();
  if (t == 0) {              // softmax over 4 logits
    const float mx = fmaxf(fmaxf(slog[0], slog[1]), fmaxf(slog[2], slog[3]));
    const float e0 = __expf(slog[0] - mx), e1 = __expf(slog[1] - mx);
    const float e2 = __expf(slog[2] - mx), e3 = __expf(slog[3] - mx);
    const float inv = 1.f / (e0 + e1 + e2 + e3);
    out[0] = e0 * inv; out[1] = e1 * inv; out[2] = e2 * inv; out[3] = e3 * inv;
  }
}

// ---------------------------------------------------------------------------
extern "C" void kernel_launch(void* const* d_in, const int* in_sizes, int n_in,
                              void* d_out, int out_size, void* d_ws, size_t ws_size,
                              hipStream_t stream) {
  const float* state    = (const float*)d_in[0];
  const float* p_action = (const float*)d_in[1];
  const float* p_reward = (const float*)d_in[2];
  const float* timestep = (const float*)d_in[3];
  const float* cue      = (const float*)d_in[4];
  const float* mem_keys = (const float*)d_in[5];
  const float* mem_vals = (const float*)d_in[6];
  const float* Wx       = (const float*)d_in[7];
  const float* Wh       = (const float*)d_in[8];
  const float* bias     = (const float*)d_in[9];
  const float* actor_w  = (const float*)d_in[10];
  const float* actor_b  = (const float*)d_in[11];
  const float* critic_w = (const float*)d_in[12];
  const float* critic_b = (const float*)d_in[13];
  const float* h0       = (const float*)d_in[14];
  const float* c0       = (const float*)d_in[15];
  float* out = (float*)d_out;

  const int N = in_sizes[5] / HDIM;  // 100000 memory rows

  // Workspace layout (floats)
  float* w = (float*)d_ws;
  float* ws_sims    = w;                                  // N
  float* ws_bmax    = ws_sims + N;                        // G1
  float* ws_gmax    = ws_bmax + G1;                       // 1
  float* ws_partial = ws_gmax + 1;                        // G3*(H+1)
  float* ws_m       = ws_partial + (size_t)G3 * (HDIM+1); // H+1 (m_unnorm, denom)
  float* ws_preact  = ws_m + (HDIM + 1);                  // 5H

  // DND scan (critical path, HBM-bound)
  k_sims<<<G1, BLK, 0, stream>>>(mem_keys, cue, ws_sims, ws_bmax, N);
  k_maxreduce<<<1, BLK, 0, stream>>>(ws_bmax, ws_gmax, G1);
  k_vals<<<G3, BLK, 0, stream>>>(mem_vals, ws_sims, ws_gmax, ws_partial, N);
  k_mreduce<<<1, BLK, 0, stream>>>(ws_partial, ws_m, G3);

  // LSTM preactivations (WMMA matvec) — independent of DND result
  k_preact<<<(NROWS5H / 16 + WAVES - 1) / WAVES, BLK, 0, stream>>>(
      Wh, Wx, bias, h0, state, p_action, p_reward, timestep, ws_preact);

  // Gates + heads
  k_final<<<1, HDIM, 0, stream>>>(ws_preact, ws_m, c0, actor_w, actor_b,
                                  critic_w, critic_b, out);
}